// CenterLoss_88699664597343
// MI455X (gfx1250) — compile-verified
//
#include <hip/hip_runtime.h>

typedef __attribute__((ext_vector_type(2))) float v2f;
typedef __attribute__((ext_vector_type(4))) float v4f;
typedef __attribute__((ext_vector_type(8))) float v8f;

#define FEAT 2048
#define NCLS 751
#define BATCH 8192
#define ROWS_PER_BLK 16
#define KSPAN 256   // K range per wave (8 waves * 256 = 2048)

// ---------------- kernel 1: csq[c] = ||centers[c]||^2 ----------------
__global__ __launch_bounds__(256) void cl_csq_kernel(const float* __restrict__ centers,
                                                     float* __restrict__ csq) {
    const int c = blockIdx.x;
    const float* row = centers + (size_t)c * FEAT;
    float s = 0.f;
    for (int k = threadIdx.x * 4; k < FEAT; k += 256 * 4) {
        v4f v = *(const v4f*)(row + k);
        s += v[0]*v[0] + v[1]*v[1] + v[2]*v[2] + v[3]*v[3];
    }
    for (int off = 16; off; off >>= 1) s += __shfl_xor(s, off, 32);
    __shared__ float sw[8];
    if ((threadIdx.x & 31) == 0) sw[threadIdx.x >> 5] = s;
    __syncthreads();
    if (threadIdx.x == 0) {
        float t = 0.f;
        #pragma unroll
        for (int i = 0; i < 8; ++i) t += sw[i];
        csq[c] = t;
    }
}

// ---------------- kernel 2: per-block masked distances via WMMA ----------------
// Block = 16 rows of x. Wave w covers K in [w*256, w*256+256).
// A tile: lane L<16 -> row L, K={k,k+1}; lane L+16 -> row L, K={k+4,k+5} (K-order free).
// B tile: same mapping with centers[label[row]] -> D[i][j] = partial dot(x_i, c_lab_j).
__global__ __launch_bounds__(256) void cl_main_kernel(const float* __restrict__ x,
                                                      const int* __restrict__ labels,
                                                      const float* __restrict__ centers,
                                                      const float* __restrict__ csq,
                                                      float* __restrict__ partials) {
    __shared__ float s_diag[ROWS_PER_BLK];
    __shared__ float s_xsq[ROWS_PER_BLK];
    if (threadIdx.x < ROWS_PER_BLK) { s_diag[threadIdx.x] = 0.f; s_xsq[threadIdx.x] = 0.f; }
    __syncthreads();

    const int lane = threadIdx.x & 31;
    const int wave = threadIdx.x >> 5;
    const int half = lane >> 4;            // 0: K-low pair, 1: K-high pair
    const int mrow = lane & 15;            // row within tile (M for A, N for B)
    const int grow = blockIdx.x * ROWS_PER_BLK + mrow;
    const int lab  = labels[grow];
    const float* __restrict__ xrow = x + (size_t)grow * FEAT;
    const float* __restrict__ crow = centers + (size_t)lab * FEAT;

    v8f acc = {0.f, 0.f, 0.f, 0.f, 0.f, 0.f, 0.f, 0.f};
    float xsq = 0.f;

    const int kbeg = wave * KSPAN;
    #pragma unroll 4
    for (int kb = kbeg; kb < kbeg + KSPAN; kb += 8) {
        const int ko = kb + half * 4;
        v4f xa = *(const v4f*)(xrow + ko);   // 16B contiguous per lane
        v4f cb = *(const v4f*)(crow + ko);
        v2f a0 = {xa[0], xa[1]}, a1 = {xa[2], xa[3]};
        v2f b0 = {cb[0], cb[1]}, b1 = {cb[2], cb[3]};
        // D = A(16x4,f32) * B(4x16,f32) + D   (full f32 precision)
        acc = __builtin_amdgcn_wmma_f32_16x16x4_f32(false, a0, false, b0,
                                                    (short)0, acc, false, false);
        acc = __builtin_amdgcn_wmma_f32_16x16x4_f32(false, a1, false, b1,
                                                    (short)0, acc, false, false);
        xsq += xa[0]*xa[0] + xa[1]*xa[1] + xa[2]*xa[2] + xa[3]*xa[3];
    }

    // pair lanes (L, L^16) together cover this wave's whole K range for row mrow
    xsq += __shfl_xor(xsq, 16, 32);

    // diagonal D[i][i]: i<8 -> lane i, VGPR i ; i>=8 -> lane i+16, VGPR i-8
    const bool isDiag = (lane < 8) || (lane >= 24);
    int idx = ((lane < 8) ? lane : (lane - 24)) & 7;
    float dv = acc[0];
    #pragma unroll
    for (int j = 1; j < 8; ++j) dv = (idx == j) ? acc[j] : dv;

    if (isDiag) atomicAdd(&s_diag[(lane < 8) ? lane : (lane - 16)], dv);
    if (lane < 16) atomicAdd(&s_xsq[lane], xsq);
    __syncthreads();

    // finalize 16 rows in wave 0, deterministic per-block partial
    float v = 0.f;
    if (threadIdx.x < ROWS_PER_BLK) {
        const int r  = blockIdx.x * ROWS_PER_BLK + threadIdx.x;
        const int lb = labels[r];
        float dist = s_xsq[threadIdx.x] + csq[lb] - 2.0f * s_diag[threadIdx.x];
        v = fminf(fmaxf(dist, 1e-12f), 1e12f);
    }
    if (threadIdx.x < 32) {
        for (int off = 16; off; off >>= 1) v += __shfl_xor(v, off, 32);
        if (threadIdx.x == 0) partials[blockIdx.x] = v;
    }
}

// ---------------- kernel 3: reduce partials, add clamp constant, divide ----------------
__global__ __launch_bounds__(256) void cl_final_kernel(const float* __restrict__ partials,
                                                       float* __restrict__ out) {
    float s = 0.f;
    for (int i = threadIdx.x; i < BATCH / ROWS_PER_BLK; i += 256) s += partials[i];
    for (int off = 16; off; off >>= 1) s += __shfl_xor(s, off, 32);
    __shared__ float sw[8];
    if ((threadIdx.x & 31) == 0) sw[threadIdx.x >> 5] = s;
    __syncthreads();
    if (threadIdx.x == 0) {
        float t = 0.f;
        #pragma unroll
        for (int i = 0; i < 8; ++i) t += sw[i];
        // every masked-out entry clamps 0 -> 1e-12: B*(C-1) of them
        t += (float)BATCH * (float)(NCLS - 1) * 1e-12f;
        out[0] = t / (float)BATCH;
    }
}

extern "C" void kernel_launch(void* const* d_in, const int* in_sizes, int n_in,
                              void* d_out, int out_size, void* d_ws, size_t ws_size,
                              hipStream_t stream) {
    const float* x       = (const float*)d_in[0];
    const int*   labels  = (const int*)d_in[1];
    const float* centers = (const float*)d_in[2];
    float* out = (float*)d_out;

    float* ws       = (float*)d_ws;
    float* csq      = ws;          // NCLS floats
    float* partials = ws + 768;    // BATCH/16 = 512 floats

    cl_csq_kernel  <<<NCLS, 256, 0, stream>>>(centers, csq);
    cl_main_kernel <<<BATCH / ROWS_PER_BLK, 256, 0, stream>>>(x, labels, centers, csq, partials);
    cl_final_kernel<<<1, 256, 0, stream>>>(partials, out);
}